// Supernetwork_21208548508281
// MI455X (gfx1250) — compile-verified
//
#include <hip/hip_runtime.h>
#include <hip/hip_bf16.h>

typedef __attribute__((ext_vector_type(2))) float v2f;
typedef __attribute__((ext_vector_type(8))) float v8f;

#define HID   128
#define BEAMS 4
#define NOPS  4
#define DEPTH 3
#define NB    2048
#define COUT  32

// ---------------------------------------------------------------------------
// Implicit-GEMM conv (VALID, 3x3) using V_WMMA_F32_16X16X4_F32.
// M = B*Hout*Wout output positions, N = 32 output channels.  One wave computes
// BOTH 16-wide N-tiles so the expensive A-patch gathers are loaded once and
// fed to two back-to-back WMMAs (A-operand reuse), K = Cin*9 padded to x4.
// A (16x4 f32): lanes 0-15 hold {K=k0,k0+1} for M=lane; lanes 16-31 hold
// {K=k0+2,k0+3} for M=lane-16.  B (4x16) mirrored with N in place of M.
// C/D (16x16 f32): VGPR r -> M=r (lanes 0-15) / M=r+8 (lanes 16-31), N=lane&15.
// ---------------------------------------------------------------------------
__global__ __launch_bounds__(128) void conv_wmma(
    const float* __restrict__ in, const float* __restrict__ w,
    const float* __restrict__ bias, float* __restrict__ out,
    int Cin, int Hin, int Win, int Hout, int Wout, int M, int K, int Ksteps)
{
    const int lane  = threadIdx.x & 31;
    const int wave  = threadIdx.x >> 5;
    const int tileM = blockIdx.x * 4 + wave;
    if (tileM * 16 >= M) return;

    const int half = lane >> 4;
    const int lm   = lane & 15;
    const int HW   = Hout * Wout;

    // A-row coordinates for this lane (clamped; OOB rows never stored)
    int m  = tileM * 16 + lm;
    int mc = (m < M) ? m : (M - 1);
    int b  = mc / HW;
    int r  = mc - b * HW;
    int oh = r / Wout;
    int ow = r - oh * Wout;
    const float* abase = in + ((size_t)b * Cin * Hin + oh) * Win + ow;

    // B-columns for this lane: co = lm (tile 0) and co = 16+lm (tile 1)
    const float* wb0 = w + (size_t)lm        * Cin * 9;
    const float* wb1 = w + (size_t)(16 + lm) * Cin * 9;

    v8f acc0 = {}, acc1 = {};
    for (int ks = 0; ks < Ksteps; ++ks) {
        const int k0 = ks * 4 + half * 2;
        v2f av, bv0, bv1;
        #pragma unroll
        for (int t = 0; t < 2; ++t) {
            const int k = k0 + t;
            float a = 0.f, b0 = 0.f, b1 = 0.f;
            if (k < K) {
                int ci  = k / 9;
                int rem = k - ci * 9;
                int kh  = rem / 3;
                int kw  = rem - kh * 3;
                a  = abase[((size_t)ci * Hin + kh) * Win + kw];
                b0 = wb0[k];
                b1 = wb1[k];
            }
            if (t == 0) { av.x = a; bv0.x = b0; bv1.x = b1; }
            else        { av.y = a; bv0.y = b0; bv1.y = b1; }
        }
        acc0 = __builtin_amdgcn_wmma_f32_16x16x4_f32(
                   false, av, false, bv0, (short)0, acc0, false, false);
        acc1 = __builtin_amdgcn_wmma_f32_16x16x4_f32(
                   false, av, false, bv1, (short)0, acc1, false, false);
    }

    const float bs0 = bias[lm];
    const float bs1 = bias[16 + lm];
    #pragma unroll
    for (int rr = 0; rr < 8; ++rr) {
        const int row = half ? rr + 8 : rr;
        const int mo  = tileM * 16 + row;
        if (mo < M) {
            int bb  = mo / HW;
            int r2  = mo - bb * HW;
            int oh2 = r2 / Wout;
            int ow2 = r2 - oh2 * Wout;
            const size_t base = (((size_t)bb * COUT) * Hout + oh2) * Wout + ow2;
            out[base + (size_t)lm        * HW] = acc0[rr] + bs0;
            out[base + (size_t)(16 + lm) * HW] = acc1[rr] + bs1;
        }
    }
}

// Per-channel sum / sum-of-squares reduction (for batch-norm statistics).
__global__ __launch_bounds__(256) void stats_kernel(
    const float* __restrict__ x, float* __restrict__ stats, int HW, int N)
{
    const int c = blockIdx.y;
    __shared__ float ssum[256], ssq[256];
    const size_t per = (size_t)N * HW;
    float s = 0.f, q = 0.f;
    for (size_t i = (size_t)blockIdx.x * blockDim.x + threadIdx.x; i < per;
         i += (size_t)gridDim.x * blockDim.x) {
        size_t n  = i / HW;
        size_t hw = i - n * HW;
        float v = x[(n * COUT + c) * HW + hw];
        s += v; q += v * v;
    }
    ssum[threadIdx.x] = s; ssq[threadIdx.x] = q;
    __syncthreads();
    for (int o = 128; o > 0; o >>= 1) {
        if (threadIdx.x < (unsigned)o) {
            ssum[threadIdx.x] += ssum[threadIdx.x + o];
            ssq[threadIdx.x]  += ssq[threadIdx.x + o];
        }
        __syncthreads();
    }
    if (threadIdx.x == 0) {
        atomicAdd(&stats[c],        ssum[0]);
        atomicAdd(&stats[COUT + c], ssq[0]);
    }
}

// Fused batch-norm + relu + 2x2 average pool (floor-cropped).
__global__ __launch_bounds__(256) void bn_relu_pool(
    const float* __restrict__ x, const float* __restrict__ stats,
    const float* __restrict__ g, const float* __restrict__ be,
    float* __restrict__ out, int H, int W, int PH, int PW, int N)
{
    size_t idx = (size_t)blockIdx.x * blockDim.x + threadIdx.x;
    const size_t total = (size_t)N * COUT * PH * PW;
    if (idx >= total) return;
    int pw = idx % PW; size_t t = idx / PW;
    int ph = t % PH; t /= PH;
    int c  = t % COUT;
    int n  = (int)(t / COUT);

    const float cnt = (float)N * (float)H * (float)W;
    const float mu  = stats[c] / cnt;
    const float var = stats[COUT + c] / cnt - mu * mu;
    const float sc  = rsqrtf(var + 1e-5f) * g[c];
    const float sh  = be[c] - mu * sc;

    const float* base = x + (((size_t)n * COUT + c) * H + 2 * ph) * W + 2 * pw;
    float a  = fmaxf(fmaf(base[0],    sc, sh), 0.f);
    float d1 = fmaxf(fmaf(base[1],    sc, sh), 0.f);
    float d2 = fmaxf(fmaf(base[W],    sc, sh), 0.f);
    float d3 = fmaxf(fmaf(base[W+1],  sc, sh), 0.f);
    out[idx] = 0.25f * (a + d1 + d2 + d3);
}

// Beam search: one 128-thread block per batch element.  Stores per
// (depth, n, beam): op index and selected probability p.
__global__ __launch_bounds__(HID) void beam_kernel(
    const float* __restrict__ h0g, const float* __restrict__ W_hh,
    const float* __restrict__ b_h, const float* __restrict__ E_op,
    const float* __restrict__ W_ho, const float* __restrict__ b_ho,
    int* __restrict__ oi, float* __restrict__ pv)
{
    const int n = blockIdx.x;
    const int j = threadIdx.x;
    __shared__ float h0s[HID];
    __shared__ float hcur[BEAMS][HID];
    __shared__ float hnext[BEAMS][HID];
    __shared__ float lg[BEAMS * NOPS];
    __shared__ float scores[BEAMS];
    __shared__ int   srcS[BEAMS], opS[BEAMS];
    __shared__ float pS[BEAMS];

    h0s[j] = h0g[(size_t)n * HID + j];
    __syncthreads();

    // step 0: logits over NOPS from h0
    if (j < NOPS) {
        float s = b_ho[j];
        for (int i = 0; i < HID; ++i) s += h0s[i] * W_ho[i * NOPS + j];
        lg[j] = s;
    }
    __syncthreads();
    if (j == 0) {
        float mx = lg[0];
        for (int o = 1; o < NOPS; ++o) mx = fmaxf(mx, lg[o]);
        float ex[NOPS], se = 0.f;
        for (int o = 0; o < NOPS; ++o) { ex[o] = __expf(lg[o] - mx); se += ex[o]; }
        float lse = mx + __logf(se);
        float lp[NOPS];
        for (int o = 0; o < NOPS; ++o) lp[o] = lg[o] - lse;
        bool used[NOPS] = {false, false, false, false};
        for (int bm = 0; bm < BEAMS; ++bm) {
            int best = 0; float bv = -3.0e38f;
            for (int o = 0; o < NOPS; ++o)
                if (!used[o] && lp[o] > bv) { bv = lp[o]; best = o; }
            used[best] = true;
            scores[bm] = bv; opS[bm] = best; pS[bm] = ex[best] / se;
        }
    }
    __syncthreads();
    if (j < BEAMS) {
        oi[((size_t)0 * NB + n) * BEAMS + j] = opS[j];
        pv[((size_t)0 * NB + n) * BEAMS + j] = pS[j];
    }
    // h = tanh(h0 @ W_hh + E_op[op] + b_h); h0@W_hh shared across beams
    float hw0 = b_h[j];
    for (int i = 0; i < HID; ++i) hw0 += h0s[i] * W_hh[i * HID + j];
    for (int bm = 0; bm < BEAMS; ++bm)
        hcur[bm][j] = tanhf(hw0 + E_op[opS[bm] * HID + j]);
    __syncthreads();

    for (int d = 1; d < DEPTH; ++d) {
        if (j < BEAMS * NOPS) {
            const int bm = j >> 2, o = j & 3;
            float s = b_ho[o];
            for (int i = 0; i < HID; ++i) s += hcur[bm][i] * W_ho[i * NOPS + o];
            lg[j] = s;
        }
        __syncthreads();
        if (j == 0) {
            float cand[BEAMS * NOPS], pr[BEAMS][NOPS];
            for (int bm = 0; bm < BEAMS; ++bm) {
                float mx = lg[bm * 4];
                for (int o = 1; o < NOPS; ++o) mx = fmaxf(mx, lg[bm * 4 + o]);
                float ex[NOPS], se = 0.f;
                for (int o = 0; o < NOPS; ++o) { ex[o] = __expf(lg[bm * 4 + o] - mx); se += ex[o]; }
                float lse = mx + __logf(se);
                for (int o = 0; o < NOPS; ++o) {
                    cand[bm * 4 + o] = scores[bm] + (lg[bm * 4 + o] - lse);
                    pr[bm][o] = ex[o] / se;
                }
            }
            bool used[BEAMS * NOPS] = {};
            float nsc[BEAMS], np[BEAMS]; int nsrc[BEAMS], nop[BEAMS];
            for (int bm = 0; bm < BEAMS; ++bm) {
                int best = 0; float bv = -3.0e38f;
                for (int t2 = 0; t2 < BEAMS * NOPS; ++t2)
                    if (!used[t2] && cand[t2] > bv) { bv = cand[t2]; best = t2; }
                used[best] = true;
                nsc[bm] = bv; nsrc[bm] = best >> 2; nop[bm] = best & 3;
                np[bm]  = pr[best >> 2][best & 3];
            }
            for (int bm = 0; bm < BEAMS; ++bm) {
                scores[bm] = nsc[bm]; srcS[bm] = nsrc[bm];
                opS[bm] = nop[bm];    pS[bm]   = np[bm];
            }
        }
        __syncthreads();
        if (j < BEAMS) {
            oi[((size_t)d * NB + n) * BEAMS + j] = opS[j];
            pv[((size_t)d * NB + n) * BEAMS + j] = pS[j];
        }
        for (int bm = 0; bm < BEAMS; ++bm) {
            float s = b_h[j];
            const float* hs = hcur[srcS[bm]];
            for (int i = 0; i < HID; ++i) s += hs[i] * W_hh[i * HID + j];
            hnext[bm][j] = tanhf(s + E_op[opS[bm] * HID + j]);
        }
        __syncthreads();
        for (int bm = 0; bm < BEAMS; ++bm) hcur[bm][j] = hnext[bm][j];
        __syncthreads();
    }
}

// Broadcast h0 (2048x128) to hid (8192x128).
__global__ __launch_bounds__(256) void init_hid(
    const float* __restrict__ h0, float* __restrict__ hid)
{
    size_t idx = (size_t)blockIdx.x * blockDim.x + threadIdx.x;
    if (idx >= (size_t)NB * BEAMS * HID) return;
    int dcol = idx % HID;
    size_t m = idx / HID;
    size_t n = m / BEAMS;
    hid[idx] = h0[n * HID + dcol];
}

// One MoE-dispatch round via WMMA: compute all 3 experts for a 16x16 tile,
// select per output row by (op, p), apply relu(sel * p).
__global__ __launch_bounds__(128) void dispatch_wmma(
    const float* __restrict__ hid_in, const float* __restrict__ exp_w,
    const float* __restrict__ exp_b, const int* __restrict__ oi,
    const float* __restrict__ pv, float* __restrict__ hid_out, int d)
{
    const int lane   = threadIdx.x & 31;
    const int wave   = threadIdx.x >> 5;
    const int gwid   = blockIdx.x * 4 + wave;
    const int tilesN = HID / 16;                 // 8
    const int tileN  = gwid % tilesN;
    const int tileM  = gwid / tilesN;
    const int half   = lane >> 4;
    const int lm     = lane & 15;

    const int mrow = tileM * 16 + lm;            // A row for this lane
    const int col  = tileN * 16 + lm;            // B/D column

    v8f acc0 = {}, acc1 = {}, acc2 = {};
    const float* arow = hid_in + (size_t)mrow * HID;
    __builtin_prefetch(arow, 0, 3);
    __builtin_prefetch(exp_w + col, 0, 3);
    #pragma unroll 4
    for (int ks = 0; ks < HID / 4; ++ks) {
        const int k0 = ks * 4 + half * 2;
        v2f a, b0, b1, b2;
        a.x  = arow[k0];
        a.y  = arow[k0 + 1];
        b0.x = exp_w[((size_t)0 * HID + k0)     * HID + col];
        b0.y = exp_w[((size_t)0 * HID + k0 + 1) * HID + col];
        b1.x = exp_w[((size_t)1 * HID + k0)     * HID + col];
        b1.y = exp_w[((size_t)1 * HID + k0 + 1) * HID + col];
        b2.x = exp_w[((size_t)2 * HID + k0)     * HID + col];
        b2.y = exp_w[((size_t)2 * HID + k0 + 1) * HID + col];
        acc0 = __builtin_amdgcn_wmma_f32_16x16x4_f32(false, a, false, b0, (short)0, acc0, false, false);
        acc1 = __builtin_amdgcn_wmma_f32_16x16x4_f32(false, a, false, b1, (short)0, acc1, false, false);
        acc2 = __builtin_amdgcn_wmma_f32_16x16x4_f32(false, a, false, b2, (short)0, acc2, false, false);
    }

    #pragma unroll
    for (int rr = 0; rr < 8; ++rr) {
        const int row = half ? rr + 8 : rr;
        const int m   = tileM * 16 + row;
        const int nn  = m >> 2;
        const int bm  = m & 3;
        const int op  = oi[((size_t)d * NB + nn) * BEAMS + bm];
        const float p = pv[((size_t)d * NB + nn) * BEAMS + bm];
        float val;
        if      (op == 0) val = acc0[rr] + exp_b[0 * HID + col];
        else if (op == 1) val = acc1[rr] + exp_b[1 * HID + col];
        else if (op == 2) val = acc2[rr] + exp_b[2 * HID + col];
        else              val = hid_in[(size_t)m * HID + col];
        hid_out[(size_t)m * HID + col] = fmaxf(val * p, 0.f);
    }
}

// Final projection hid (8192x128) @ out_w (128x10) + out_b.
__global__ __launch_bounds__(256) void final_out(
    const float* __restrict__ hid, const float* __restrict__ ow,
    const float* __restrict__ ob, float* __restrict__ out)
{
    const int OC = 10;
    int idx = blockIdx.x * blockDim.x + threadIdx.x;
    if (idx >= NB * BEAMS * OC) return;
    int c = idx % OC, m = idx / OC;
    float s = ob[c];
    for (int i = 0; i < HID; ++i) s += hid[(size_t)m * HID + i] * ow[i * OC + c];
    out[idx] = s;
}

extern "C" void kernel_launch(void* const* d_in, const int* in_sizes, int n_in,
                              void* d_out, int out_size, void* d_ws, size_t ws_size,
                              hipStream_t stream)
{
    const float* x    = (const float*)d_in[0];
    const float* cw1  = (const float*)d_in[1];
    const float* cb1  = (const float*)d_in[2];
    const float* g1   = (const float*)d_in[3];
    const float* be1  = (const float*)d_in[4];
    const float* cw2  = (const float*)d_in[5];
    const float* cb2  = (const float*)d_in[6];
    const float* g2   = (const float*)d_in[7];
    const float* be2  = (const float*)d_in[8];
    const float* cw3  = (const float*)d_in[9];
    const float* cb3  = (const float*)d_in[10];
    const float* g3   = (const float*)d_in[11];
    const float* be3  = (const float*)d_in[12];
    const float* W_hh = (const float*)d_in[13];
    const float* b_h  = (const float*)d_in[14];
    const float* E_op = (const float*)d_in[15];
    const float* W_ho = (const float*)d_in[16];
    const float* b_ho = (const float*)d_in[17];
    const float* exp_w = (const float*)d_in[18];
    const float* exp_b = (const float*)d_in[19];
    const float* out_w = (const float*)d_in[20];
    const float* out_b = (const float*)d_in[21];
    float* out = (float*)d_out;

    // ---- workspace layout (256B aligned) ----
    char* base = (char*)d_ws;
    size_t off = 0;
    auto alloc = [&](size_t bytes) -> char* {
        char* p = base + off;
        off += (bytes + 255) & ~(size_t)255;
        return p;
    };
    float* A     = (float*)alloc((size_t)NB * COUT * 30 * 30 * 4); // conv scratch (reused layers 1-3)
    float* P1    = (float*)alloc((size_t)NB * COUT * 15 * 15 * 4);
    float* P2    = (float*)alloc((size_t)NB * COUT * 6 * 6 * 4);
    float* H0    = (float*)alloc((size_t)NB * HID * 4);
    float* HIDA  = (float*)alloc((size_t)NB * BEAMS * HID * 4);
    float* HIDB  = (float*)alloc((size_t)NB * BEAMS * HID * 4);
    float* stats = (float*)alloc(3 * 64 * 4);                      // [layer][sum(32)|sq(32)]
    int*   oi    = (int*)  alloc((size_t)DEPTH * NB * BEAMS * 4);
    float* pv    = (float*)alloc((size_t)DEPTH * NB * BEAMS * 4);
    (void)ws_size; (void)n_in; (void)in_sizes; (void)out_size;

    hipMemsetAsync(stats, 0, 3 * 64 * sizeof(float), stream);

    // ---- layer 1: conv(3->32, 32x32 -> 30x30), BN, relu, pool -> 15x15 ----
    {
        const int M = NB * 30 * 30;
        const int blocks = (M / 16 + 3) / 4;
        conv_wmma<<<blocks, 128, 0, stream>>>(x, cw1, cb1, A, 3, 32, 32, 30, 30, M, 27, 7);
        stats_kernel<<<dim3(240, COUT), 256, 0, stream>>>(A, stats + 0 * 64, 900, NB);
        const size_t tot = (size_t)NB * COUT * 15 * 15;
        bn_relu_pool<<<(tot + 255) / 256, 256, 0, stream>>>(A, stats + 0 * 64, g1, be1, P1, 30, 30, 15, 15, NB);
    }
    // ---- layer 2: conv(32->32, 15x15 -> 13x13), BN, relu, pool -> 6x6 ----
    {
        const int M = NB * 13 * 13;
        const int blocks = (M / 16 + 3) / 4;
        conv_wmma<<<blocks, 128, 0, stream>>>(P1, cw2, cb2, A, 32, 15, 15, 13, 13, M, 288, 72);
        stats_kernel<<<dim3(240, COUT), 256, 0, stream>>>(A, stats + 1 * 64, 169, NB);
        const size_t tot = (size_t)NB * COUT * 6 * 6;
        bn_relu_pool<<<(tot + 255) / 256, 256, 0, stream>>>(A, stats + 1 * 64, g2, be2, P2, 13, 13, 6, 6, NB);
    }
    // ---- layer 3: conv(32->32, 6x6 -> 4x4), BN, relu, pool -> 2x2 == h0 ----
    {
        const int M = NB * 4 * 4;
        const int blocks = (M / 16 + 3) / 4;
        conv_wmma<<<blocks, 128, 0, stream>>>(P2, cw3, cb3, A, 32, 6, 6, 4, 4, M, 288, 72);
        stats_kernel<<<dim3(64, COUT), 256, 0, stream>>>(A, stats + 2 * 64, 16, NB);
        const size_t tot = (size_t)NB * COUT * 2 * 2;   // (n,32,2,2) contiguous == (n,128)
        bn_relu_pool<<<(tot + 255) / 256, 256, 0, stream>>>(A, stats + 2 * 64, g3, be3, H0, 4, 4, 2, 2, NB);
    }

    // ---- beam search ----
    beam_kernel<<<NB, HID, 0, stream>>>(H0, W_hh, b_h, E_op, W_ho, b_ho, oi, pv);

    // ---- dispatch (3 rounds, ping-pong) ----
    {
        const size_t tot = (size_t)NB * BEAMS * HID;
        init_hid<<<(tot + 255) / 256, 256, 0, stream>>>(H0, HIDA);
        const int M = NB * BEAMS;                       // 8192
        const int blocks = ((M / 16) * (HID / 16)) / 4; // 1024
        dispatch_wmma<<<blocks, 128, 0, stream>>>(HIDA, exp_w, exp_b, oi, pv, HIDB, 0);
        dispatch_wmma<<<blocks, 128, 0, stream>>>(HIDB, exp_w, exp_b, oi, pv, HIDA, 1);
        dispatch_wmma<<<blocks, 128, 0, stream>>>(HIDA, exp_w, exp_b, oi, pv, HIDB, 2);
    }

    // ---- final projection ----
    {
        const int tot = NB * BEAMS * 10;
        final_out<<<(tot + 255) / 256, 256, 0, stream>>>(HIDB, out_w, out_b, out);
    }
}